// Joint_58050777973339
// MI455X (gfx1250) — compile-verified
//
#include <hip/hip_runtime.h>
#include <hip/hip_bf16.h>

#define UPS     512
#define NPIX    (UPS*UPS)        // 262144
#define BB      16
#define RADIUS  60
#define DD      (2*RADIUS)       // 120
#define NPATCH  (DD*DD)          // 14400
#define COEF    1.5f
#define DOT2    16               // dot radius^2

typedef float v2f __attribute__((ext_vector_type(2)));
typedef float v8f __attribute__((ext_vector_type(8)));

// ---------------------------------------------------------------------------
// Per-wave affine transform of 32 pixel coordinates via V_WMMA_F32_16X16X4_F32.
// D = A(16x4) x B(4x16):
//   A row0 = [t00 t01 t02 0], row1 = [t10 t11 t12 0]
//   B col n = [X_n Y_n 1 0]
// D[0,n] = gx_n, D[1,n] = gy_n land in lanes 0-15 VGPR0/1.  Two WMMAs cover
// the 32 lanes; ds_bpermute (__shfl) moves batch-1 results to lanes 16-31.
// Requires: all 32 lanes active, all lanes of a wave share the same theta.
// ---------------------------------------------------------------------------
__device__ __forceinline__ void affine32(float X, float Y,
                                         const float* __restrict__ t,
                                         float& gx, float& gy)
{
    const int  lane = threadIdx.x & 31;
    const int  m    = lane & 15;
    const bool lo   = lane < 16;
    const float t00 = t[0], t01 = t[1], t02 = t[2];
    const float t10 = t[3], t11 = t[4], t12 = t[5];

    v2f a;
    a.x = (m == 0) ? (lo ? t00 : t02) : ((m == 1) ? (lo ? t10 : t12) : 0.0f);
    a.y = lo ? ((m == 0) ? t01 : ((m == 1) ? t11 : 0.0f)) : 0.0f;

    v2f b0; b0.x = lo ? X : 1.0f; b0.y = lo ? Y : 0.0f;     // pixels 0..15
    float Xh = __shfl(X, m + 16, 32);
    float Yh = __shfl(Y, m + 16, 32);
    v2f b1; b1.x = lo ? Xh : 1.0f; b1.y = lo ? Yh : 0.0f;   // pixels 16..31

    v8f c = {};
    v8f d0 = __builtin_amdgcn_wmma_f32_16x16x4_f32(false, a, false, b0,
                                                   (short)0, c, false, false);
    v8f d1 = __builtin_amdgcn_wmma_f32_16x16x4_f32(false, a, false, b1,
                                                   (short)0, c, false, false);
    float gxh = __shfl(d1[0], m, 32);
    float gyh = __shfl(d1[1], m, 32);
    gx = lo ? d0[0] : gxh;
    gy = lo ? d0[1] : gyh;
}

// Branchless bilinear sample with zero padding: clamp gather indices, zero the
// weight of out-of-range taps (exactly torch/jax grid_sample zero-pad math,
// which also gathers at clipped indices and zeroes the weight).
__device__ __forceinline__ float bilin_zero(const float* __restrict__ im,
                                            int H, int W, float x, float y)
{
    float xf = floorf(x), yf = floorf(y);
    float wx = x - xf,  wy = y - yf;
    int x0 = (int)xf, y0 = (int)yf;
    int x1 = x0 + 1,  y1 = y0 + 1;
    bool vx0 = (unsigned)x0 < (unsigned)W, vx1 = (unsigned)x1 < (unsigned)W;
    bool vy0 = (unsigned)y0 < (unsigned)H, vy1 = (unsigned)y1 < (unsigned)H;
    int cx0 = min(max(x0, 0), W - 1), cx1 = min(max(x1, 0), W - 1);
    int cy0 = min(max(y0, 0), H - 1), cy1 = min(max(y1, 0), H - 1);
    float w00 = (1.0f - wx) * (1.0f - wy), w10 = wx * (1.0f - wy);
    float w01 = (1.0f - wx) * wy,          w11 = wx * wy;
    // same tap order as the reference: (x0,y0),(x0+1,y0),(x0,y0+1),(x0+1,y0+1)
    return im[cy0 * W + cx0] * ((vx0 & vy0) ? w00 : 0.0f)
         + im[cy0 * W + cx1] * ((vx1 & vy0) ? w10 : 0.0f)
         + im[cy1 * W + cx0] * ((vx0 & vy1) ? w01 : 0.0f)
         + im[cy1 * W + cx1] * ((vx1 & vy1) ? w11 : 0.0f);
}

// Inverses of the three 2x3 affines (append [0,0,1], invert, keep top 2 rows).
__global__ void k_inverse(const float* __restrict__ sc,
                          const float* __restrict__ rot,
                          const float* __restrict__ tr,
                          float* __restrict__ invOut)
{
    int t = threadIdx.x;
    if (t >= 3 * BB) return;
    int which = t / BB, b = t % BB;
    const float* src = (which == 0) ? sc : (which == 1) ? rot : tr;
    const float* th = src + b * 6;
    float a = th[0], bb = th[1], c = th[2], d = th[3], e = th[4], f = th[5];
    float inv = 1.0f / (a * e - bb * d);
    float* o = invOut + which * (BB * 6) + b * 6;
    o[0] =  e * inv; o[1] = -bb * inv; o[2] = (bb * f - c * e) * inv;
    o[3] = -d * inv; o[4] =  a * inv;  o[5] = (c * d - a * f) * inv;
}

// jax.image.resize 'linear' 128->512: half-pixel centers, clamped taps.
__global__ void k_resize(const float* __restrict__ src, float* __restrict__ dst)
{
    unsigned idx = blockIdx.x * blockDim.x + threadIdx.x;   // BB*NPIX threads
    int b = idx >> 18;
    int pix = idx & (NPIX - 1);
    int oy = pix >> 9, ox = pix & (UPS - 1);
    float sy = (oy + 0.5f) * 0.25f - 0.5f;
    float sx = (ox + 0.5f) * 0.25f - 0.5f;
    float yf = floorf(sy), xf = floorf(sx);
    float fy = sy - yf, fx = sx - xf;
    int y0 = (int)yf, x0 = (int)xf;
    int y0c = min(max(y0, 0), 127),     y1c = min(max(y0 + 1, 0), 127);
    int x0c = min(max(x0, 0), 127),     x1c = min(max(x0 + 1, 0), 127);
    const float* im = src + (size_t)b * 16384;
    dst[idx] = (1.0f - fy) * ((1.0f - fx) * im[y0c * 128 + x0c] + fx * im[y0c * 128 + x1c])
             +         fy  * ((1.0f - fx) * im[y1c * 128 + x0c] + fx * im[y1c * 128 + x1c]);
}

// One affine grid_sample pass over (BB,1,512,512).
__global__ void k_gsample(const float* __restrict__ src, float* __restrict__ dst,
                          const float* __restrict__ th)
{
    unsigned idx = blockIdx.x * blockDim.x + threadIdx.x;   // BB*NPIX threads
    int b = idx >> 18;
    int pix = idx & (NPIX - 1);
    int i = pix >> 9, jj = pix & (UPS - 1);
    const float* t = th + b * 6;
    float X = (2 * jj + 1) * (1.0f / UPS) - 1.0f;
    float Y = (2 * i  + 1) * (1.0f / UPS) - 1.0f;
    float gx, gy; affine32(X, Y, t, gx, gy);
    float x = ((gx + 1.0f) * UPS - 1.0f) * 0.5f;
    float y = ((gy + 1.0f) * UPS - 1.0f) * 0.5f;
    dst[idx] = bilin_zero(src + (size_t)b * NPIX, UPS, UPS, x, y);
}

// m2(ix,iy) = grid_sample(mask_j, grid(it2)) evaluated at one integer pixel.
__device__ __forceinline__ float m2_at(const float* __restrict__ mj,
                                       const float* __restrict__ t2,
                                       int ix, int iy)
{
    float X2 = (2 * ix + 1) * (1.0f / UPS) - 1.0f;
    float Y2 = (2 * iy + 1) * (1.0f / UPS) - 1.0f;
    float hx = t2[0] * X2 + t2[1] * Y2 + t2[2];
    float hy = t2[3] * X2 + t2[4] * Y2 + t2[5];
    float xx = ((hx + 1.0f) * UPS - 1.0f) * 0.5f;
    float yy = ((hy + 1.0f) * UPS - 1.0f) * 0.5f;
    return bilin_zero(mj, UPS, UPS, xx, yy);
}

// Fused gs(gs(mask, g2), g1) >= 0.5 -> 0/1 float.  4 outer taps (branchless:
// clamped coords, zeroed weights) x 4 inner taps each.
__global__ void k_masks(const float* __restrict__ masks, float* __restrict__ outD,
                        const float* __restrict__ it1, const float* __restrict__ it2)
{
    unsigned idx = blockIdx.x * blockDim.x + threadIdx.x;   // 4*BB*NPIX threads
    int j = idx >> 22;
    int b = (idx >> 18) & 15;
    int pix = idx & (NPIX - 1);
    int i = pix >> 9, jj = pix & (UPS - 1);
    const float* t1 = it1 + b * 6;
    const float* t2 = it2 + b * 6;
    const float* mj = masks + (size_t)j * NPIX;
    float X = (2 * jj + 1) * (1.0f / UPS) - 1.0f;
    float Y = (2 * i  + 1) * (1.0f / UPS) - 1.0f;
    float gx, gy; affine32(X, Y, t1, gx, gy);
    float x = ((gx + 1.0f) * UPS - 1.0f) * 0.5f;
    float y = ((gy + 1.0f) * UPS - 1.0f) * 0.5f;
    float xf = floorf(x), yf = floorf(y);
    float wx = x - xf, wy = y - yf;
    int x0 = (int)xf, y0 = (int)yf;
    int x1 = x0 + 1,  y1 = y0 + 1;
    bool vx0 = (unsigned)x0 < UPS, vx1 = (unsigned)x1 < UPS;
    bool vy0 = (unsigned)y0 < UPS, vy1 = (unsigned)y1 < UPS;
    int cx0 = min(max(x0, 0), UPS - 1), cx1 = min(max(x1, 0), UPS - 1);
    int cy0 = min(max(y0, 0), UPS - 1), cy1 = min(max(y1, 0), UPS - 1);
    float acc = m2_at(mj, t2, cx0, cy0) * ((vx0 & vy0) ? (1.0f - wx) * (1.0f - wy) : 0.0f)
              + m2_at(mj, t2, cx1, cy0) * ((vx1 & vy0) ? wx * (1.0f - wy) : 0.0f)
              + m2_at(mj, t2, cx0, cy1) * ((vx0 & vy1) ? (1.0f - wx) * wy : 0.0f)
              + m2_at(mj, t2, cx1, cy1) * ((vx1 & vy1) ? wx * wy : 0.0f);
    outD[idx] = (acc >= 0.5f) ? 1.0f : 0.0f;
}

// Center-of-mass: one block per batch item, two-phase reduction.
__global__ void k_com(const float* __restrict__ pred,
                      const float* __restrict__ maskj,
                      int* __restrict__ comXY)
{
    int b = blockIdx.x, tid = threadIdx.x;
    const float* img = pred  + (size_t)b * NPIX;
    const float* m   = maskj + (size_t)b * NPIX;
    __shared__ float s0[256], s1[256], s2[256];
    __shared__ float thr_s;

    float sm = 0.0f, sn = 0.0f;
    for (int p = tid; p < NPIX; p += 256) { float mm = m[p]; sm += mm; sn += img[p] * mm; }
    s0[tid] = sm; s1[tid] = sn; __syncthreads();
    for (int off = 128; off; off >>= 1) {
        if (tid < off) { s0[tid] += s0[tid + off]; s1[tid] += s1[tid + off]; }
        __syncthreads();
    }
    if (tid == 0) thr_s = (s1[0] / fmaxf(s0[0], 1.0f)) * COEF;
    __syncthreads();
    float thr = thr_s;

    float sw = 0.0f, swr = 0.0f, swc = 0.0f;
    for (int p = tid; p < NPIX; p += 256) {
        float v = img[p] * m[p];
        if (v > thr) { sw += v; swr += v * (float)(p >> 9); swc += v * (float)(p & (UPS - 1)); }
    }
    s0[tid] = sw; s1[tid] = swr; s2[tid] = swc; __syncthreads();
    for (int off = 128; off; off >>= 1) {
        if (tid < off) { s0[tid] += s0[tid + off]; s1[tid] += s1[tid + off]; s2[tid] += s2[tid + off]; }
        __syncthreads();
    }
    if (tid == 0) {
        float tot = s0[0] + 1e-8f;
        float cx = s1[0] / tot, cy = s2[0] / tot;          // row, col centroid
        int x0 = min(max((int)rintf(cx) - RADIUS, 0), UPS - DD);   // dynamic_slice clamp
        int y0 = min(max((int)rintf(cy) - RADIUS, 0), UPS - DD);
        comXY[2 * b] = x0; comXY[2 * b + 1] = y0;
    }
}

// Extract 120x120 patch, divide dot circle by adj (pare_reverse=True).
__global__ void k_extract(const float* __restrict__ imgC,
                          const int* __restrict__ comXY,
                          const float* __restrict__ adj,
                          float* __restrict__ patch)
{
    int b = blockIdx.y;
    int pix = blockIdx.x * blockDim.x + threadIdx.x;
    if (pix >= NPATCH) return;
    int r = pix / DD, c = pix % DD;
    int x0 = comXY[2 * b], y0 = comXY[2 * b + 1];
    float v = imgC[(size_t)b * NPIX + (size_t)(x0 + r) * UPS + (y0 + c)];
    int dr = r - RADIUS, dc = c - RADIUS;
    if (dr * dr + dc * dc <= DOT2) v /= adj[b];
    patch[(size_t)b * NPATCH + pix] = v;
}

// Resample patch with it1's 120x120 affine grid, write back in place.
// No early-return before the WMMA (EXEC must be all ones); store is predicated.
__global__ void k_scatter(const float* __restrict__ patch,
                          const int* __restrict__ comXY,
                          const float* __restrict__ it1,
                          float* __restrict__ imgC)
{
    int b = blockIdx.y;
    int pix0 = blockIdx.x * blockDim.x + threadIdx.x;
    int pix = min(pix0, NPATCH - 1);
    int r = pix / DD, c = pix % DD;
    const float* t = it1 + b * 6;
    float X = (2 * c + 1) * (1.0f / DD) - 1.0f;
    float Y = (2 * r + 1) * (1.0f / DD) - 1.0f;
    float gx, gy; affine32(X, Y, t, gx, gy);
    float x = ((gx + 1.0f) * DD - 1.0f) * 0.5f;
    float y = ((gy + 1.0f) * DD - 1.0f) * 0.5f;
    float v = bilin_zero(patch + (size_t)b * NPATCH, DD, DD, x, y);
    if (pix0 < NPATCH) {
        int x0 = comXY[2 * b], y0 = comXY[2 * b + 1];
        imgC[(size_t)b * NPIX + (size_t)(x0 + r) * UPS + (y0 + c)] = v;
    }
}

extern "C" void kernel_launch(void* const* d_in, const int* in_sizes, int n_in,
                              void* d_out, int out_size, void* d_ws, size_t ws_size,
                              hipStream_t stream)
{
    (void)in_sizes; (void)n_in; (void)out_size; (void)ws_size;
    const float* pb    = (const float*)d_in[0];   // (16,1,128,128)
    const float* sc    = (const float*)d_in[1];   // (16,2,3)
    const float* rot   = (const float*)d_in[2];   // (16,2,3)
    const float* tr    = (const float*)d_in[3];   // (16,2,3)
    const float* adj   = (const float*)d_in[4];   // (16,)
    const float* masks = (const float*)d_in[5];   // (4,512,512)

    float* out  = (float*)d_out;
    float* outA = out;                               // base_inp
    float* outB = out + (size_t)1 * BB * NPIX;       // pred_input
    float* outC = out + (size_t)2 * BB * NPIX;       // pred_revise
    float* outD = out + (size_t)3 * BB * NPIX;       // masks_rot (also scratch early)

    float* wsF   = (float*)d_ws;
    float* it1   = wsF;            // 96 floats
    float* it2   = wsF + 96;
    float* it3   = wsF + 192;
    int*   comXY = (int*)(wsF + 288);   // 32 ints
    float* patch = wsF + 320;           // 16*14400 floats (~0.9 MB)

    const int nblk  = (BB * NPIX) / 256;        // 16384
    const int nblkM = (4 * BB * NPIX) / 256;    // 65536

    k_inverse<<<1, 64, 0, stream>>>(sc, rot, tr, wsF);
    k_resize <<<nblk, 256, 0, stream>>>(pb, outA);
    k_gsample<<<nblk, 256, 0, stream>>>(outA, outC, it3);  // pred_trans -> C (scratch)
    k_gsample<<<nblk, 256, 0, stream>>>(outC, outD, it2);  // pred_rot   -> D (scratch)
    k_gsample<<<nblk, 256, 0, stream>>>(outD, outB, it1);  // pred_input -> B
    k_masks  <<<nblkM, 256, 0, stream>>>(masks, outD, it1, it2); // masks_rot -> D
    hipMemcpyAsync(outC, outB, (size_t)BB * NPIX * sizeof(float),
                   hipMemcpyDeviceToDevice, stream);             // revise seed

    dim3 gp((NPATCH + 255) / 256, BB);
    for (int j = 0; j < 4; ++j) {
        const float* maskj = outD + (size_t)j * BB * NPIX;
        k_com    <<<BB, 256, 0, stream>>>(outB, maskj, comXY);
        k_extract<<<gp, 256, 0, stream>>>(outC, comXY, adj, patch);
        k_scatter<<<gp, 256, 0, stream>>>(patch, comXY, it1, outC);
    }
}